// NeuralMemoryManager_17051020165242
// MI455X (gfx1250) — compile-verified
//
#include <hip/hip_runtime.h>
#include <math.h>

// Problem constants (match reference)
#define NB   128
#define DIM  512
#define HID  1024

__constant__ const float kLR  = 0.01f;
__constant__ const float kWD  = 0.01f;
__constant__ const float kEPS = 1e-8f;

typedef __attribute__((ext_vector_type(2))) float v2f;
typedef __attribute__((ext_vector_type(4))) float v4f;
typedef __attribute__((ext_vector_type(8))) float v8f;

// ---------------------------------------------------------------------------
// Helpers
// ---------------------------------------------------------------------------
__device__ __forceinline__ float gelu_f(float z) {
  // exact GELU (erf), matches jax.nn.gelu(approximate=False) / torch default
  return 0.5f * z * (1.0f + erff(z * 0.70710678118654752f));
}
__device__ __forceinline__ float dgelu_f(float z) {
  float cdf = 0.5f * (1.0f + erff(z * 0.70710678118654752f));
  float pdf = 0.3989422804014327f * expf(-0.5f * z * z);
  return cdf + z * pdf;
}
// step-1 AdamW collapses to: p*(1-lr*wd) - lr * g/(|g|+eps)
__device__ __forceinline__ float adamw1(float p, float g, float& wch2) {
  float pn = p * (1.0f - kLR * kWD) - kLR * (g / (fabsf(g) + kEPS));
  float d  = pn - p;
  wch2 += d * d;
  return pn;
}

template <int N>
__device__ __forceinline__ float block_reduce(float v, float* red) {
  const int tid = threadIdx.x;
  red[tid] = v;
  __syncthreads();
#pragma unroll
  for (int s = N / 2; s > 0; s >>= 1) {
    if (tid < s) red[tid] += red[tid + s];
    __syncthreads();
  }
  float r = red[0];
  __syncthreads();
  return r;
}

// ---------------------------------------------------------------------------
// K0: V = X @ Wv + bv  (128x512x512 fp32 GEMM) via V_WMMA_F32_16X16X4_F32.
// One wave per 16x16 output tile. Lane layouts per CDNA5 ISA 7.12.2:
//   A 16x4 f32 : lanes 0-15 {K=0,K=1}, lanes 16-31 {K=2,K=3}, M = lane&15
//   B 4x16 f32 : lanes 0-15 rows {K=0,K=1}, lanes 16-31 rows {K=2,K=3}, N = lane&15
//   C/D 16x16  : VGPR r -> M = r + 8*(lane>=16), N = lane&15
// ---------------------------------------------------------------------------
__global__ __launch_bounds__(32) void value_target_wmma(
    const float* __restrict__ X, const float* __restrict__ Wv,
    const float* __restrict__ bv, float* __restrict__ V) {
  const int lane = threadIdx.x & 31;
  const int half = lane >> 4;
  const int l    = lane & 15;
  const int n0   = blockIdx.x * 16;  // over DIM (512)
  const int m0   = blockIdx.y * 16;  // over NB (128)

  v8f c = {};
  const float* __restrict__ xr = X + (size_t)(m0 + l) * DIM;
  for (int k0 = 0; k0 < DIM; k0 += 4) {
    v2f a, b;
    const int ka = k0 + 2 * half;
    a.x = xr[ka];
    a.y = xr[ka + 1];
    b.x = Wv[(size_t)ka * DIM + n0 + l];
    b.y = Wv[(size_t)(ka + 1) * DIM + n0 + l];
    c = __builtin_amdgcn_wmma_f32_16x16x4_f32(
        /*neg_a=*/false, a, /*neg_b=*/false, b,
        /*c_mod=*/(short)0, c, /*reuse_a=*/false, /*reuse_b=*/false);
  }
  const float bvn = bv[n0 + l];
#pragma unroll
  for (int r = 0; r < 8; ++r)
    V[(size_t)(m0 + r + 8 * half) * DIM + n0 + l] = c[r] + bvn;
}

// ---------------------------------------------------------------------------
// K1: per-bot forward + backward vectors + norms + bias updates.
// One 512-thread block per bot. Streams W1 once (coalesced float2 columns)
// and W2 twice (second pass is L2-resident). First-touch reads keep default
// RT temporal hint so the update kernels' re-reads hit L2.
// ---------------------------------------------------------------------------
__global__ __launch_bounds__(512) void forward_kernel(
    const float* __restrict__ x, const float* __restrict__ W1,
    const float* __restrict__ b1, const float* __restrict__ W2,
    const float* __restrict__ b2, const float* __restrict__ V,
    float* __restrict__ o_retr, float* __restrict__ o_loss,
    float* __restrict__ o_b1, float* __restrict__ o_b2,
    float* __restrict__ ws_h, float* __restrict__ ws_dz,
    float* __restrict__ ws_dp, float* __restrict__ ws_coef,
    float* __restrict__ ws_part) {
  const int bot = blockIdx.x;
  const int tid = threadIdx.x;

  __shared__ float sx[DIM];
  __shared__ float sz[HID];   // z (each element read back only by its owner)
  __shared__ float sh[HID];   // gelu(z)
  __shared__ float sdp[DIM];  // d_pred
  __shared__ float red[512];

  const float* __restrict__ W1b = W1 + (size_t)bot * DIM * HID;
  const float* __restrict__ W2b = W2 + (size_t)bot * HID * DIM;

  const float xv = x[(size_t)bot * DIM + tid];
  sx[tid] = xv;
  __syncthreads();

  // ---- z = x@W1 + b1 ; h = gelu(z); thread owns cols j = 2t, 2t+1 ----
  const int j0 = 2 * tid;
  v2f acc1;
  acc1.x = b1[(size_t)bot * HID + j0];
  acc1.y = b1[(size_t)bot * HID + j0 + 1];
  for (int i = 0; i < DIM; ++i) {
    const v2f w = *(const v2f*)(W1b + (size_t)i * HID + j0);
    const float xi = sx[i];
    acc1.x = fmaf(xi, w.x, acc1.x);
    acc1.y = fmaf(xi, w.y, acc1.y);
  }
  sz[j0]     = acc1.x;
  sz[j0 + 1] = acc1.y;
  const float h0 = gelu_f(acc1.x), h1 = gelu_f(acc1.y);
  sh[j0]     = h0;
  sh[j0 + 1] = h1;
  __syncthreads();

  // ---- pred = h@W2 + b2 ; d_pred ; thread owns col n = tid ----
  float acc2 = b2[(size_t)bot * DIM + tid];
  for (int j = 0; j < HID; ++j)
    acc2 = fmaf(sh[j], W2b[(size_t)j * DIM + tid], acc2);
  const float vtg = V[(size_t)bot * DIM + tid];
  const float dif = acc2 - vtg;
  o_retr[(size_t)bot * DIM + tid] = acc2;
  const float dpn = dif * (2.0f / (float)DIM);
  sdp[tid] = dpn;

  const float S_loss = block_reduce<512>(dif * dif, red);  // also syncs sdp

  // ---- dh = W2 @ d_pred ; dz = dh * gelu'(z); thread owns rows 2t,2t+1 ----
  float dzv[2];
#pragma unroll
  for (int r = 0; r < 2; ++r) {
    const int j = j0 + r;
    const v4f* __restrict__ row = (const v4f*)(W2b + (size_t)j * DIM);
    float s = 0.0f;
    for (int n4 = 0; n4 < DIM / 4; ++n4) {
      const v4f w = row[n4];
      s = fmaf(w.x, sdp[4 * n4 + 0], s);
      s = fmaf(w.y, sdp[4 * n4 + 1], s);
      s = fmaf(w.z, sdp[4 * n4 + 2], s);
      s = fmaf(w.w, sdp[4 * n4 + 3], s);
    }
    dzv[r] = s * dgelu_f(sz[j]);
  }

  // ---- grad norm via rank-1 identity: ||x⊗dz||_F = ||x||·||dz|| ----
  const float S_x2  = block_reduce<512>(xv * xv, red);
  const float S_dz2 = block_reduce<512>(dzv[0] * dzv[0] + dzv[1] * dzv[1], red);
  const float S_h2  = block_reduce<512>(h0 * h0 + h1 * h1, red);
  const float S_dp2 = block_reduce<512>(dpn * dpn, red);

  const float gnorm = sqrtf(S_x2 * S_dz2 + S_dz2 + S_h2 * S_dp2 + S_dp2);
  const float coef  = fminf(1.0f, 1.0f / (gnorm + 1e-6f));

  // ---- bias updates (fused step-1 AdamW) + partial ||Δ||² ----
  float wch2 = 0.0f;
#pragma unroll
  for (int r = 0; r < 2; ++r) {
    const int j   = j0 + r;
    const float p = b1[(size_t)bot * HID + j];
    o_b1[(size_t)bot * HID + j] = adamw1(p, coef * dzv[r], wch2);
  }
  {
    const float p = b2[(size_t)bot * DIM + tid];
    o_b2[(size_t)bot * DIM + tid] = adamw1(p, coef * dpn, wch2);
  }
  const float S_wch = block_reduce<512>(wch2, red);

  // ---- stash vectors for the weight-update kernels ----
  ws_h[(size_t)bot * HID + j0]      = h0;
  ws_h[(size_t)bot * HID + j0 + 1]  = h1;
  ws_dz[(size_t)bot * HID + j0]     = dzv[0];
  ws_dz[(size_t)bot * HID + j0 + 1] = dzv[1];
  ws_dp[(size_t)bot * DIM + tid]    = dpn;
  if (tid == 0) {
    o_loss[bot]  = S_loss / (float)DIM;
    ws_coef[bot] = coef;
    ws_part[bot * 17 + 16] = S_wch;
  }
}

// ---------------------------------------------------------------------------
// K2: W1' update. grid (8 chunks, NB bots), 256 threads. Each block streams
// 64 rows x 1024 cols; g = coef * x[i] * dz[j] (never materialized as a grad).
// Last-use NT loads of W1; NT stores of W1' (never re-read -> don't pollute
// the 192MB L2 that is keeping the other bots' W1/W2 resident).
// ---------------------------------------------------------------------------
__global__ __launch_bounds__(256) void update_w1(
    const float* __restrict__ x, const float* __restrict__ W1,
    const float* __restrict__ ws_dz, const float* __restrict__ ws_coef,
    float* __restrict__ o_W1, float* __restrict__ ws_part) {
  const int bot = blockIdx.y, chunk = blockIdx.x, tid = threadIdx.x;
  __shared__ float sxc[64];
  __shared__ float red[256];
  if (tid < 64) sxc[tid] = x[(size_t)bot * DIM + chunk * 64 + tid];
  const float coef = ws_coef[bot];
  __syncthreads();

  const v4f dz4 = *(const v4f*)(ws_dz + (size_t)bot * HID + tid * 4);
  const size_t base =
      (size_t)bot * DIM * HID + (size_t)chunk * 64 * HID + (size_t)tid * 4;

  float wch2 = 0.0f;
  for (int r = 0; r < 64; ++r) {
    const float cg = coef * sxc[r];
    const v4f w =
        __builtin_nontemporal_load((const v4f*)(W1 + base + (size_t)r * HID));
    v4f nw;
    nw.x = adamw1(w.x, cg * dz4.x, wch2);
    nw.y = adamw1(w.y, cg * dz4.y, wch2);
    nw.z = adamw1(w.z, cg * dz4.z, wch2);
    nw.w = adamw1(w.w, cg * dz4.w, wch2);
    __builtin_nontemporal_store(nw, (v4f*)(o_W1 + base + (size_t)r * HID));
  }
  const float S = block_reduce<256>(wch2, red);
  if (tid == 0) ws_part[bot * 17 + chunk] = S;
}

// ---------------------------------------------------------------------------
// K3: W2' update. grid (8 chunks, NB bots), 256 threads. 128 rows x 512 cols
// per block; g = coef * h[j] * d_pred[n]. NT last-use loads / NT stores.
// ---------------------------------------------------------------------------
__global__ __launch_bounds__(256) void update_w2(
    const float* __restrict__ W2, const float* __restrict__ ws_h,
    const float* __restrict__ ws_dp, const float* __restrict__ ws_coef,
    float* __restrict__ o_W2, float* __restrict__ ws_part) {
  const int bot = blockIdx.y, chunk = blockIdx.x, tid = threadIdx.x;
  __shared__ float shc[128];
  __shared__ float red[256];
  if (tid < 128) shc[tid] = ws_h[(size_t)bot * HID + chunk * 128 + tid];
  const float coef = ws_coef[bot];
  __syncthreads();

  const v2f dp2 = *(const v2f*)(ws_dp + (size_t)bot * DIM + tid * 2);
  const size_t base =
      (size_t)bot * HID * DIM + (size_t)chunk * 128 * DIM + (size_t)tid * 2;

  float wch2 = 0.0f;
  for (int r = 0; r < 128; ++r) {
    const float cg = coef * shc[r];
    const v2f w =
        __builtin_nontemporal_load((const v2f*)(W2 + base + (size_t)r * DIM));
    v2f nw;
    nw.x = adamw1(w.x, cg * dp2.x, wch2);
    nw.y = adamw1(w.y, cg * dp2.y, wch2);
    __builtin_nontemporal_store(nw, (v2f*)(o_W2 + base + (size_t)r * DIM));
  }
  const float S = block_reduce<256>(wch2, red);
  if (tid == 0) ws_part[bot * 17 + 8 + chunk] = S;
}

// ---------------------------------------------------------------------------
// K4: finalize wchange = sqrt(sum of 17 deterministic partials per bot).
// ---------------------------------------------------------------------------
__global__ __launch_bounds__(NB) void finalize_kernel(
    const float* __restrict__ ws_part, float* __restrict__ o_wch) {
  const int bot = threadIdx.x;
  float s = 0.0f;
#pragma unroll
  for (int k = 0; k < 17; ++k) s += ws_part[bot * 17 + k];
  o_wch[bot] = sqrtf(s);
}

// ---------------------------------------------------------------------------
// Host launcher
// ---------------------------------------------------------------------------
extern "C" void kernel_launch(void* const* d_in, const int* in_sizes, int n_in,
                              void* d_out, int out_size, void* d_ws,
                              size_t ws_size, hipStream_t stream) {
  (void)in_sizes; (void)n_in; (void)out_size; (void)ws_size;
  const float* x  = (const float*)d_in[0];
  const float* W1 = (const float*)d_in[1];
  const float* b1 = (const float*)d_in[2];
  const float* W2 = (const float*)d_in[3];
  const float* b2 = (const float*)d_in[4];
  const float* Wv = (const float*)d_in[5];
  const float* bv = (const float*)d_in[6];

  float* out = (float*)d_out;
  float* o_retr = out;                                   // [NB, DIM]
  float* o_loss = o_retr + (size_t)NB * DIM;             // [NB]
  float* o_wch  = o_loss + NB;                           // [NB]
  float* o_W1   = o_wch + NB;                            // [NB, DIM, HID]
  float* o_b1   = o_W1 + (size_t)NB * DIM * HID;         // [NB, HID]
  float* o_W2   = o_b1 + (size_t)NB * HID;               // [NB, HID, DIM]
  float* o_b2   = o_W2 + (size_t)NB * HID * DIM;         // [NB, DIM]

  float* ws      = (float*)d_ws;
  float* w_V     = ws;                                   // [NB, DIM]
  float* w_h     = w_V + (size_t)NB * DIM;               // [NB, HID]
  float* w_dz    = w_h + (size_t)NB * HID;               // [NB, HID]
  float* w_dp    = w_dz + (size_t)NB * HID;              // [NB, DIM]
  float* w_coef  = w_dp + (size_t)NB * DIM;              // [NB]
  float* w_part  = w_coef + NB;                          // [NB, 17]

  // K0: shared value-target GEMM via fp32 WMMA
  value_target_wmma<<<dim3(DIM / 16, NB / 16), 32, 0, stream>>>(x, Wv, bv, w_V);
  // K1: per-bot forward/backward vectors, norms, bias updates
  forward_kernel<<<NB, 512, 0, stream>>>(x, W1, b1, W2, b2, w_V, o_retr,
                                         o_loss, o_b1, o_b2, w_h, w_dz, w_dp,
                                         w_coef, w_part);
  // K2/K3: streamed weight updates with fused step-1 AdamW (NT traffic)
  update_w1<<<dim3(8, NB), 256, 0, stream>>>(x, W1, w_dz, w_coef, o_W1, w_part);
  update_w2<<<dim3(8, NB), 256, 0, stream>>>(W2, w_h, w_dp, w_coef, o_W2, w_part);
  // K4: wchange
  finalize_kernel<<<1, NB, 0, stream>>>(w_part, o_wch);
}